// IDWT_13580686590053
// MI455X (gfx1250) — compile-verified
//
#include <hip/hip_runtime.h>

// IDWT (inverse 2x2 Haar): x:[16, 4*64, 128, 128] f32 -> out:[16, 64, 256, 256] f32.
// Memory-bound (AI ~= 1 FLOP/byte, ~537 MB traffic -> ~23 us floor @ 23.3 TB/s).
// Per 16-pixel group the einsum is a 4x4 * 4x16 f32 matmul -> V_WMMA_F32_16X16X4_F32.
// A-matrix rows replicate with period 4, so D's lanes 16-31 duplicate lanes 0-15:
// one full-wave b64 store covers BOTH output rows (row 2h from lanes 0-15 via
// d[0..1], row 2h+1 from lanes 16-31 via d[2..3]) with no EXEC masking.

typedef __attribute__((ext_vector_type(2))) float v2f;
typedef __attribute__((ext_vector_type(8))) float v8f;

namespace {
constexpr int kB = 16, kC = 64, kH = 128, kW = 128;
constexpr long long kSB = (long long)kC * kH * kW;  // subband stride (elements)
constexpr int kTPB = 256;                           // 8 waves/block
constexpr int kBlocks = 2048;                       // 16384 waves total
// Each wave handles 8 h-rows x 8 w-groups = 64 groups of 16 pixels.
// 16384 waves * 64 groups = 1,048,576 = 16*64*128*8 groups. Exact cover.
}  // namespace

__global__ __launch_bounds__(kTPB) void idwt_haar_wmma(
    const float* __restrict__ x, const float* __restrict__ f,
    float* __restrict__ out) {
  const int lane = threadIdx.x & 31;
  const int hf   = lane >> 4;   // 0: subbands {0,1}; 1: subbands {2,3}
  const int m    = lane & 15;   // A-matrix row / pixel column
  const int wid  = blockIdx.x * (kTPB / 32) + (threadIdx.x >> 5);

  // A (16x4 f32): vgpr0 = K={0|2}, vgpr1 = K={1|3}; A[M][K] = f[K*4 + (M&3)].
  // Rows repeat with period 4 -> D rows 8+r duplicate rows r (used by the store).
  const int pq = m & 3;
  v2f a;
  a.x = f[(2 * hf + 0) * 4 + pq];
  a.y = f[(2 * hf + 1) * 4 + pq];

  // Wave -> (b, c, hBase); each wave owns h in [hBase, hBase+8), all 8 w-groups.
  const int b     = wid >> 10;
  const int c     = (wid >> 4) & 63;
  const int hBase = (wid & 15) * 8;

  const float* inP = x + ((long long)(b * 4 * kC + c)) * (kH * kW) +
                     (long long)hBase * kW + m + (long long)(2 * hf) * kSB;
  float* outP = out +
                (((long long)(b * kC + c)) * (2 * kH) + 2 * hBase) * (2 * kW) +
                hf * (2 * kW) + 2 * m;

  for (int ih = 0; ih < 8; ++ih) {
#pragma unroll
    for (int iw = 0; iw < 8; ++iw) {
      // B (4x16 f32): lane supplies pixel m of subbands (2*hf, 2*hf+1).
      v2f bv;
      bv.x = inP[iw * 16];
      bv.y = inP[iw * 16 + kSB];

      v8f acc = {};
      v8f d = __builtin_amdgcn_wmma_f32_16x16x4_f32(
          /*neg_a=*/false, a, /*neg_b=*/false, bv,
          /*c_mod=*/(short)0, acc, /*reuse_a=*/false, /*reuse_b=*/false);

      // Full-wave store: lanes 0-15 -> row 2h (out[0,0],out[0,1]),
      // lanes 16-31 -> row 2h+1 (out[1,0],out[1,1], duplicated D halves).
      v2f st;
      st.x = hf ? d[2] : d[0];
      st.y = hf ? d[3] : d[1];
      *(v2f*)(outP + iw * 32) = st;
    }
    inP += kW;            // next input row
    outP += 2 * (2 * kW); // next output row pair
  }
}

extern "C" void kernel_launch(void* const* d_in, const int* in_sizes, int n_in,
                              void* d_out, int out_size, void* d_ws, size_t ws_size,
                              hipStream_t stream) {
  const float* x = (const float*)d_in[0];   // [16, 256, 128, 128] f32
  const float* f = (const float*)d_in[1];   // [4, 2, 2] f32
  float* out = (float*)d_out;               // [16, 64, 256, 256] f32
  (void)in_sizes; (void)n_in; (void)out_size; (void)d_ws; (void)ws_size;
  idwt_haar_wmma<<<kBlocks, kTPB, 0, stream>>>(x, f, out);
}